// NormActive_3891240370805
// MI455X (gfx1250) — compile-verified
//
#include <hip/hip_runtime.h>
#include <stdint.h>

#define FEATSIZE 4096
#define NACTIVE  512
#define THREADS  256
#define PER_THREAD (FEATSIZE / THREADS)   // 16 elements per thread

typedef float v4f __attribute__((ext_vector_type(4)));

// Map float bits to a monotonically ordered unsigned key:
// larger float  <=>  larger key (handles negative values and +/-0).
__device__ __forceinline__ uint32_t f2key(uint32_t b) {
    return b ^ ((b & 0x80000000u) ? 0xFFFFFFFFu : 0x80000000u);
}

__global__ __launch_bounds__(THREADS)
void NormActive_topk_kernel(const float* __restrict__ feat,
                            float* __restrict__ out,
                            int batch)
{
    __shared__ uint32_t s_row[FEATSIZE];   // raw row bits (16 KB)
    __shared__ uint32_t s_hist[256];
    __shared__ uint32_t s_scan[256];
    __shared__ uint32_t s_prefix;          // selected key prefix (threshold being built)
    __shared__ uint32_t s_k;               // how many still needed from current bin
    __shared__ uint32_t s_tie;             // tie-break counter for == threshold

    const int row = blockIdx.x;
    if (row >= batch) return;              // grid == batch, uniform per block
    const int tid = threadIdx.x;

    const float* __restrict__ src = feat + (size_t)row * FEATSIZE;
    float*       __restrict__ dst = out  + (size_t)row * FEATSIZE;

    // ---- Stage 1: async DMA row -> LDS (CDNA5 async-to-LDS path) ----------
    // Each thread issues 4x b128 loads; consecutive threads touch consecutive
    // 16B chunks -> fully coalesced. Tracked on ASYNCcnt, not LOADcnt.
    {
        uint32_t lds_base = (uint32_t)(uintptr_t)(&s_row[0]);
#pragma unroll
        for (int c = 0; c < 4; ++c) {
            uint32_t elem   = (uint32_t)(c * 1024 + tid * 4);
            uint32_t ldsoff = lds_base + elem * 4u;
            uint32_t goff   = elem * 4u;
            asm volatile("global_load_async_to_lds_b128 %0, %1, %2 offset:0"
                         :
                         : "v"(ldsoff), "v"(goff), "s"(src)
                         : "memory");
        }
        asm volatile("s_wait_asynccnt 0" ::: "memory");
    }
    if (tid == 0) { s_prefix = 0u; s_k = NACTIVE; s_tie = 0u; }
    __syncthreads();

    // ---- Stage 2: radix-select the NACTIVE-th largest key in LDS ----------
    // 4 passes over byte digits, MSB first. Candidates = elements whose
    // higher bytes match the prefix chosen so far.
#pragma unroll
    for (int p = 3; p >= 0; --p) {
        s_hist[tid] = 0u;
        __syncthreads();

        const uint32_t prefix = s_prefix;
        const int      shift  = 8 * p;
        const uint32_t pmask  = (p == 3) ? 0u : (0xFFFFFFFFu << (shift + 8));

#pragma unroll
        for (int j = 0; j < PER_THREAD; ++j) {
            uint32_t key = f2key(s_row[tid + j * THREADS]);
            if ((key & pmask) == prefix) {
                atomicAdd(&s_hist[(key >> shift) & 0xFFu], 1u);
            }
        }
        __syncthreads();

        // Parallel suffix sum over 256 bins: s_scan[i] = sum_{j>=i} hist[j]
        s_scan[tid] = s_hist[tid];
        __syncthreads();
        for (int off = 1; off < 256; off <<= 1) {
            uint32_t add = (tid + off < 256) ? s_scan[tid + off] : 0u;
            __syncthreads();
            s_scan[tid] += add;
            __syncthreads();
        }

        // Pick the digit d where the cumulative-from-top first reaches k.
        const uint32_t k        = s_k;
        const uint32_t cum      = s_scan[tid];
        const uint32_t cumAbove = (tid < 255) ? s_scan[tid + 1] : 0u;
        if (cum >= k && cumAbove < k) {      // exactly one thread wins
            s_prefix = prefix | ((uint32_t)tid << shift);
            s_k      = k - cumAbove;         // still needed from this bin
        }
        __syncthreads();
    }

    const uint32_t thr    = s_prefix;  // full 32-bit threshold key
    const uint32_t needEq = s_k;       // how many ==thr elements to keep

    // ---- Stage 3: scaled scatter-back with 128-bit non-temporal stores ----
    const float scale = 8.0f;          // EFF_SCALE = 4096/512
#pragma unroll
    for (int c = 0; c < 4; ++c) {
        const uint32_t elem = (uint32_t)(c * 1024 + tid * 4);
        float r[4];
#pragma unroll
        for (int q = 0; q < 4; ++q) {
            uint32_t bits = s_row[elem + q];
            uint32_t key  = f2key(bits);
            float    val  = __uint_as_float(bits);
            float    o    = 0.0f;
            if (key > thr) {
                o = val * scale;
            } else if (key == thr) {
                uint32_t slot = atomicAdd(&s_tie, 1u);
                if (slot < needEq) o = val * scale;
            }
            r[q] = o;
        }
        v4f v = { r[0], r[1], r[2], r[3] };
        __builtin_nontemporal_store(v, (v4f*)(dst + elem));
    }
}

extern "C" void kernel_launch(void* const* d_in, const int* in_sizes, int n_in,
                              void* d_out, int out_size, void* d_ws, size_t ws_size,
                              hipStream_t stream)
{
    const float* feat = (const float*)d_in[0];
    float*       out  = (float*)d_out;
    const int batch = in_sizes[0] / FEATSIZE;   // 8192

    NormActive_topk_kernel<<<batch, THREADS, 0, stream>>>(feat, out, batch);
}